// DGCNNEncoder_62354335203515
// MI455X (gfx1250) — compile-verified
//
#include <hip/hip_runtime.h>

typedef __attribute__((ext_vector_type(2))) float v2f;
typedef __attribute__((ext_vector_type(8))) float v8f;

#define WMMA4(a, b, c) \
  __builtin_amdgcn_wmma_f32_16x16x4_f32(false, (a), false, (b), (short)0, (c), false, false)

// ---------------------------------------------------------------------------
// Pairwise squared-distance tiles via fp32 WMMA.
// Block = 128 threads (4 waves). Block tile = 16 rows x 64 cols; wave w does
// the 16x16 sub-tile at col offset 16*w. K loop over channels in steps of 4.
// dist[row*N+col] = |x_row|^2 + |x_col|^2 - 2 * <x_row, x_col>
// ---------------------------------------------------------------------------
#define DIST_CMAX 128

__global__ __launch_bounds__(128) void k_dist(const float* __restrict__ x,
                                              float* __restrict__ dist,
                                              int N, int C, int b) {
  __shared__ float rowT[16][DIST_CMAX];
  __shared__ float colT[64][DIST_CMAX];
  __shared__ float xxr[16];
  __shared__ float xxc[64];
  const int row0 = blockIdx.y * 16;
  const int col0 = blockIdx.x * 64;
  const float* xb = x + (size_t)b * N * C;
  const int tid = threadIdx.x;
  const int Cpad = (C + 3) & ~3;

  for (int i = tid; i < 16 * Cpad; i += 128) {
    int m = i / Cpad, c = i % Cpad;
    rowT[m][c] = (c < C) ? xb[(size_t)(row0 + m) * C + c] : 0.f;
  }
  for (int i = tid; i < 64 * Cpad; i += 128) {
    int m = i / Cpad, c = i % Cpad;
    colT[m][c] = (c < C) ? xb[(size_t)(col0 + m) * C + c] : 0.f;
  }
  __syncthreads();
  if (tid < 16) {
    float s = 0.f;
    for (int c = 0; c < C; ++c) { float v = rowT[tid][c]; s += v * v; }
    xxr[tid] = s;
  } else if (tid < 80) {
    int m = tid - 16;
    float s = 0.f;
    for (int c = 0; c < C; ++c) { float v = colT[m][c]; s += v * v; }
    xxc[m] = s;
  }
  __syncthreads();

  const int wave = tid >> 5, lane = tid & 31;
  const int nn = lane & 15;                  // A row (M) and B col (N)
  const int koff = (lane < 16) ? 0 : 2;      // K sub-offset per half-wave
  const int half8 = (lane < 16) ? 0 : 8;     // D row offset per half-wave
  const int wc = wave * 16;

  v8f acc = {};
  for (int kk = 0; kk < Cpad; kk += 4) {
    v2f a, bb;
    a.x = rowT[nn][kk + koff];
    a.y = rowT[nn][kk + koff + 1];
    bb.x = colT[wc + nn][kk + koff];
    bb.y = colT[wc + nn][kk + koff + 1];
    acc = WMMA4(a, bb, acc);
  }

  const int col = col0 + wc + nn;
  const float xc = xxc[wc + nn];
#pragma unroll
  for (int r = 0; r < 8; ++r) {
    int m = r + half8;
    dist[(size_t)(row0 + m) * N + col] = xxr[m] + xc - 2.f * acc[r];
  }
}

// ---------------------------------------------------------------------------
// Top-k (k smallest distances) per row. Block = 4 waves, one row per wave.
// Repeated wave-wide arg-min extraction (tie-break: smaller index first,
// matching jax.lax.top_k behavior on -dist).
// ---------------------------------------------------------------------------
__global__ __launch_bounds__(128) void k_topk(const float* __restrict__ dist,
                                              int* __restrict__ idxout,
                                              int N, int K, int b) {
  __shared__ float rowbuf[4][2048];
  const int wave = threadIdx.x >> 5, lane = threadIdx.x & 31;
  const int row = blockIdx.x * 4 + wave;
  const float* dr = dist + (size_t)row * N;
  float* rb = rowbuf[wave];
  for (int i = lane; i < N; i += 32) rb[i] = dr[i];
  __syncthreads();
  int* out = idxout + (size_t)(b * N + row) * K;
  for (int t = 0; t < K; ++t) {
    float bv = 3.4e38f;
    int bi = 0;
    for (int i = lane; i < N; i += 32) {
      float v = rb[i];
      if (v < bv || (v == bv && i < bi)) { bv = v; bi = i; }
    }
#pragma unroll
    for (int off = 16; off > 0; off >>= 1) {
      float ov = __shfl_xor(bv, off);
      int oi = __shfl_xor(bi, off);
      if (ov < bv || (ov == bv && oi < bi)) { bv = ov; bi = oi; }
    }
    if (lane == 0) out[t] = bi;
    rb[bi] = 3.4e38f;  // all lanes agree on bi; same-value write is safe
    __syncthreads();
  }
}

// ---------------------------------------------------------------------------
// Edge-conv GEMM via fp32 WMMA. One block per point (b,n); LDS holds the
// 32 x 2C edge-feature tile (rows >= K zero-padded). Each wave owns 16 output
// channels per iteration; two 16-row accumulators cover the padded k dim.
// mode 0: write per-point channel sum / sumsq partials (for BN stats).
// mode 1: apply y = out*scale+shift, ReLU, max over k -> out[b,n,O].
// ---------------------------------------------------------------------------
#define EDGE_CMAX 256  // max 2C (layer 4: 2*128)

__global__ __launch_bounds__(256) void k_edge(const float* __restrict__ x,
                                              const int* __restrict__ knn,
                                              const float* __restrict__ W,
                                              float* __restrict__ esum,
                                              float* __restrict__ esumsq,
                                              const float* __restrict__ scale,
                                              const float* __restrict__ shift,
                                              float* __restrict__ out,
                                              int N, int C, int O, int K, int mode) {
  __shared__ float edge[32][EDGE_CMAX];
  const int bn = blockIdx.x;
  const int b = bn / N, n = bn % N;
  const float* xb = x + (size_t)b * N * C;
  const float* xi = xb + (size_t)n * C;
  const int* nb = knn + (size_t)bn * K;
  const int twoC = 2 * C;
  const int Kpad2 = (twoC + 3) & ~3;
  const int tid = threadIdx.x;

  for (int i = tid; i < 32 * Kpad2; i += 256) {
    int r = i / Kpad2, c = i % Kpad2;
    float v = 0.f;
    if (r < K && c < twoC) {
      if (c < C) v = xb[(size_t)nb[r] * C + c] - xi[c];
      else       v = xi[c - C];
    }
    edge[r][c] = v;
  }
  __syncthreads();

  const int wave = tid >> 5, lane = tid & 31;
  const int nn = lane & 15;
  const int koff = (lane < 16) ? 0 : 2;
  const int half8 = (lane < 16) ? 0 : 8;
  const int ogroups = O >> 4;

  for (int og = wave; og < ogroups; og += 8) {
    const float* w0 = W + (size_t)(og * 16 + nn) * twoC;  // this lane's out col
    v8f acc0 = {}, acc1 = {};
    for (int kk = 0; kk < Kpad2; kk += 4) {
      const int c0 = kk + koff;
      v2f bb, a0, a1;
      bb.x = (c0 < twoC) ? w0[c0] : 0.f;
      bb.y = (c0 + 1 < twoC) ? w0[c0 + 1] : 0.f;
      a0.x = edge[nn][c0];       a0.y = edge[nn][c0 + 1];
      a1.x = edge[16 + nn][c0];  a1.y = edge[16 + nn][c0 + 1];
      acc0 = WMMA4(a0, bb, acc0);
      acc1 = WMMA4(a1, bb, acc1);
    }
    const int colg = og * 16 + nn;
    if (mode == 0) {
      float s = 0.f, ss = 0.f;
#pragma unroll
      for (int r = 0; r < 8; ++r) {
        int row0 = r + half8;
        if (row0 < K) { float v = acc0[r]; s += v; ss += v * v; }
        int row1 = 16 + r + half8;
        if (row1 < K) { float v = acc1[r]; s += v; ss += v * v; }
      }
      s += __shfl_xor(s, 16);
      ss += __shfl_xor(ss, 16);
      if (lane < 16) {
        esum[(size_t)bn * O + colg] = s;
        esumsq[(size_t)bn * O + colg] = ss;
      }
    } else {
      const float sc = scale[colg], sh = shift[colg];
      float m = 0.f;  // ReLU outputs are >= 0, so 0 is the identity for max
#pragma unroll
      for (int r = 0; r < 8; ++r) {
        int row0 = r + half8;
        if (row0 < K) m = fmaxf(m, fmaxf(acc0[r] * sc + sh, 0.f));
        int row1 = 16 + r + half8;
        if (row1 < K) m = fmaxf(m, fmaxf(acc1[r] * sc + sh, 0.f));
      }
      m = fmaxf(m, __shfl_xor(m, 16));
      if (lane < 16) out[(size_t)bn * O + colg] = m;
    }
  }
}

// ---------------------------------------------------------------------------
// Deterministic BN-stat reduction: one block per channel; fold per-point
// partial sums -> mean/var -> precomputed scale/shift.
// ---------------------------------------------------------------------------
__global__ __launch_bounds__(256) void k_stats(const float* __restrict__ esum,
                                               const float* __restrict__ esumsq,
                                               const float* __restrict__ gamma,
                                               const float* __restrict__ beta,
                                               float* __restrict__ scale,
                                               float* __restrict__ shift,
                                               int BN, int O, float cnt) {
  __shared__ float s1[256], s2[256];
  const int o = blockIdx.x, tid = threadIdx.x;
  float a = 0.f, q = 0.f;
  for (int i = tid; i < BN; i += 256) {
    a += esum[(size_t)i * O + o];
    q += esumsq[(size_t)i * O + o];
  }
  s1[tid] = a; s2[tid] = q;
  __syncthreads();
  for (int st = 128; st > 0; st >>= 1) {
    if (tid < st) { s1[tid] += s1[tid + st]; s2[tid] += s2[tid + st]; }
    __syncthreads();
  }
  if (tid == 0) {
    float mean = s1[0] / cnt;
    float var = s2[0] / cnt - mean * mean;
    float sc = gamma[o] * rsqrtf(var + 1e-5f);
    scale[o] = sc;
    shift[o] = beta[o] - mean * sc;
  }
}

// ---------------------------------------------------------------------------
// Global pooling: per (channel, batch) block reduces max & mean over N,
// writing g = [max(feat) ; mean(feat)] with feat = concat(x1,x2,x3,x4).
// ---------------------------------------------------------------------------
__global__ __launch_bounds__(256) void k_pool(const float* __restrict__ x1,
                                              const float* __restrict__ x2,
                                              const float* __restrict__ x3,
                                              const float* __restrict__ x4,
                                              float* __restrict__ g, int N) {
  __shared__ float sM[256], sS[256];
  const int c = blockIdx.x, b = blockIdx.y, tid = threadIdx.x;
  const float* src;
  int C, cc;
  if (c < 64)       { src = x1; C = 64;  cc = c; }
  else if (c < 128) { src = x2; C = 64;  cc = c - 64; }
  else if (c < 256) { src = x3; C = 128; cc = c - 128; }
  else              { src = x4; C = 256; cc = c - 256; }
  const float* p = src + (size_t)b * N * C + cc;
  float mx = -3.4e38f, sm = 0.f;
  for (int n = tid; n < N; n += 256) {
    float v = p[(size_t)n * C];
    mx = fmaxf(mx, v);
    sm += v;
  }
  sM[tid] = mx; sS[tid] = sm;
  __syncthreads();
  for (int st = 128; st > 0; st >>= 1) {
    if (tid < st) { sM[tid] = fmaxf(sM[tid], sM[tid + st]); sS[tid] += sS[tid + st]; }
    __syncthreads();
  }
  if (tid == 0) {
    g[(size_t)b * 1024 + c] = sM[0];
    g[(size_t)b * 1024 + 512 + c] = sS[0] / (float)N;
  }
}

// ---------------------------------------------------------------------------
// MLP head: fc1(1024->512)+LN+ReLU, fc2(512->256)+LN. One block per batch row.
// Tiny (B=8), done in scalar math with block-wide LN reductions.
// ---------------------------------------------------------------------------
__global__ __launch_bounds__(512) void k_head(const float* __restrict__ g,
                                              const float* __restrict__ fc1w,
                                              const float* __restrict__ fc1b,
                                              const float* __restrict__ ln1g,
                                              const float* __restrict__ ln1b,
                                              const float* __restrict__ fc2w,
                                              const float* __restrict__ fc2b,
                                              const float* __restrict__ ln2g,
                                              const float* __restrict__ ln2b,
                                              float* __restrict__ out) {
  __shared__ float gin[1024];
  __shared__ float h[512];
  __shared__ float red[512];
  const int b = blockIdx.x, tid = threadIdx.x;
  gin[tid] = g[(size_t)b * 1024 + tid];
  gin[tid + 512] = g[(size_t)b * 1024 + 512 + tid];
  __syncthreads();

  const float* wr = fc1w + (size_t)tid * 1024;
  float acc = fc1b[tid];
  for (int j = 0; j < 1024; ++j) acc += gin[j] * wr[j];

  red[tid] = acc; __syncthreads();
  for (int st = 256; st > 0; st >>= 1) { if (tid < st) red[tid] += red[tid + st]; __syncthreads(); }
  float mu = red[0] / 512.f; __syncthreads();
  float d1 = acc - mu;
  red[tid] = d1 * d1; __syncthreads();
  for (int st = 256; st > 0; st >>= 1) { if (tid < st) red[tid] += red[tid + st]; __syncthreads(); }
  float var1 = red[0] / 512.f; __syncthreads();
  h[tid] = fmaxf(d1 * rsqrtf(var1 + 1e-5f) * ln1g[tid] + ln1b[tid], 0.f);
  __syncthreads();

  float z = 0.f;
  if (tid < 256) {
    const float* w2 = fc2w + (size_t)tid * 512;
    z = fc2b[tid];
    for (int j = 0; j < 512; ++j) z += h[j] * w2[j];
  }
  red[tid] = (tid < 256) ? z : 0.f; __syncthreads();
  for (int st = 256; st > 0; st >>= 1) { if (tid < st) red[tid] += red[tid + st]; __syncthreads(); }
  float mu2 = red[0] / 256.f; __syncthreads();
  float d2 = z - mu2;
  red[tid] = (tid < 256) ? d2 * d2 : 0.f; __syncthreads();
  for (int st = 256; st > 0; st >>= 1) { if (tid < st) red[tid] += red[tid + st]; __syncthreads(); }
  float var2 = red[0] / 256.f;
  if (tid < 256) out[(size_t)b * 256 + tid] = d2 * rsqrtf(var2 + 1e-5f) * ln2g[tid] + ln2b[tid];
}

// ---------------------------------------------------------------------------
extern "C" void kernel_launch(void* const* d_in, const int* in_sizes, int n_in,
                              void* d_out, int out_size, void* d_ws, size_t ws_size,
                              hipStream_t stream) {
  (void)in_sizes; (void)n_in; (void)out_size; (void)ws_size;
  const int B = 8, N = 2048, KNN = 20, BN = B * N;

  const float* pts = (const float*)d_in[0];
  const float* fc1w = (const float*)d_in[13];
  const float* fc1b = (const float*)d_in[14];
  const float* ln1g = (const float*)d_in[15];
  const float* ln1b = (const float*)d_in[16];
  const float* fc2w = (const float*)d_in[17];
  const float* fc2b = (const float*)d_in[18];
  const float* ln2g = (const float*)d_in[19];
  const float* ln2b = (const float*)d_in[20];
  // d_in[21] is k (device scalar) == 20; hardcoded.

  // Workspace carve-up (256B aligned).
  char* wp = (char*)d_ws;
  auto carve = [&](size_t bytes) -> void* {
    void* p = (void*)wp;
    wp += (bytes + 255) & ~(size_t)255;
    return p;
  };
  float* x1     = (float*)carve((size_t)BN * 64 * 4);
  float* x2     = (float*)carve((size_t)BN * 64 * 4);
  float* x3     = (float*)carve((size_t)BN * 128 * 4);
  float* x4     = (float*)carve((size_t)BN * 256 * 4);
  int*   idxbuf = (int*)carve((size_t)BN * KNN * 4);
  float* dist   = (float*)carve((size_t)N * N * 4);          // reused per batch
  float* esum   = (float*)carve((size_t)BN * 256 * 4);
  float* esumsq = (float*)carve((size_t)BN * 256 * 4);
  float* scbuf  = (float*)carve(256 * 4);
  float* shbuf  = (float*)carve(256 * 4);
  float* gbuf   = (float*)carve((size_t)B * 1024 * 4);

  struct Layer {
    const float* in; int C; float* out; int O;
    const float* W; const float* ga; const float* be;
  };
  const Layer L[4] = {
      {pts, 3,   x1, 64,  (const float*)d_in[1],  (const float*)d_in[2],  (const float*)d_in[3]},
      {x1,  64,  x2, 64,  (const float*)d_in[4],  (const float*)d_in[5],  (const float*)d_in[6]},
      {x2,  64,  x3, 128, (const float*)d_in[7],  (const float*)d_in[8],  (const float*)d_in[9]},
      {x3,  128, x4, 256, (const float*)d_in[10], (const float*)d_in[11], (const float*)d_in[12]},
  };

  for (int l = 0; l < 4; ++l) {
    const Layer& ly = L[l];
    for (int b = 0; b < B; ++b) {
      k_dist<<<dim3(N / 64, N / 16), 128, 0, stream>>>(ly.in, dist, N, ly.C, b);
      k_topk<<<dim3(N / 4), 128, 0, stream>>>(dist, idxbuf, N, KNN, b);
    }
    k_edge<<<dim3(BN), 256, 0, stream>>>(ly.in, idxbuf, ly.W, esum, esumsq,
                                         nullptr, nullptr, nullptr,
                                         N, ly.C, ly.O, KNN, 0);
    k_stats<<<dim3(ly.O), 256, 0, stream>>>(esum, esumsq, ly.ga, ly.be,
                                            scbuf, shbuf, BN, ly.O,
                                            (float)BN * (float)KNN);
    k_edge<<<dim3(BN), 256, 0, stream>>>(ly.in, idxbuf, ly.W, nullptr, nullptr,
                                         scbuf, shbuf, ly.out,
                                         N, ly.C, ly.O, KNN, 1);
  }

  k_pool<<<dim3(512, B), 256, 0, stream>>>(x1, x2, x3, x4, gbuf, N);
  k_head<<<dim3(B), 512, 0, stream>>>(gbuf, fc1w, fc1b, ln1g, ln1b,
                                      fc2w, fc2b, ln2g, ln2b, (float*)d_out);
}